// GraphConvBase_32839319945735
// MI455X (gfx1250) — compile-verified
//
#include <hip/hip_runtime.h>

#define N_NODES 50000
#define N_EDGES 800000
#define D 128
#define N_LAYERS 3
#define OUT_DIM 16
#define N_GRAPHS 128

typedef __attribute__((ext_vector_type(2))) float v2f;
typedef __attribute__((ext_vector_type(8))) float v8f;

// Accumulate a 16x16 f32 tile: acc += A[16 x 128] * B[128 x 16] using
// V_WMMA_F32_16X16X4_F32 (K stepped by 4). Arow = per-lane pointer to A row
// (row = tile_m*16 + (lane&15)); ascale lets us fold the mean-pool division
// into the A operand. B is row-major with leading dimension ldb; n is this
// lane's output column.
__device__ __forceinline__ v8f wmma_accum_k128(const float* __restrict__ Arow,
                                               float ascale,
                                               const float* __restrict__ B,
                                               int ldb, int n, int half,
                                               v8f acc) {
#pragma unroll
  for (int k = 0; k < D; k += 4) {
    const int ka = k + (half << 1);  // lanes 0-15: K={k,k+1}; 16-31: {k+2,k+3}
    const float2 av = *reinterpret_cast<const float2*>(Arow + ka);
    v2f a;
    a.x = av.x * ascale;
    a.y = av.y * ascale;
    v2f b;
    b.x = B[ka * ldb + n];
    b.y = B[(ka + 1) * ldb + n];
    acc = __builtin_amdgcn_wmma_f32_16x16x4_f32(false, a, false, b, (short)0,
                                                acc, false, false);
  }
  return acc;
}

// One wave (32 lanes) per edge: load h[src] row (float4/lane, 512B coalesced),
// atomically add into agg[dst]. Random rows live in the 192MB L2.
__global__ void __launch_bounds__(256) scatter_edges(
    const float* __restrict__ h, const int* __restrict__ src,
    const int* __restrict__ dst, float* __restrict__ agg) {
  const int gid = blockIdx.x * blockDim.x + threadIdx.x;
  const int e = gid >> 5;
  const int lane = gid & 31;
  if (e >= N_EDGES) return;
  const int s = src[e];
  const int d = dst[e];
  const float4 v = reinterpret_cast<const float4*>(h + (size_t)s * D)[lane];
  float* o = agg + (size_t)d * D + lane * 4;
  atomicAdd(o + 0, v.x);
  atomicAdd(o + 1, v.y);
  atomicAdd(o + 2, v.z);
  atomicAdd(o + 3, v.w);
}

// h_out = relu(agg @ Wrel + brel + h_in @ Wroot); one wave per 16x16 tile.
// grid = (2, N_NODES/16), block = (32, 4).
__global__ void __launch_bounds__(128) graphconv_gemm(
    const float* __restrict__ agg, const float* __restrict__ hin,
    const float* __restrict__ Wrel, const float* __restrict__ brel,
    const float* __restrict__ Wroot, float* __restrict__ hout) {
  const int lane = threadIdx.x;
  const int half = lane >> 4;
  const int l16 = lane & 15;
  const int tile_n = threadIdx.y + blockDim.y * blockIdx.x;  // 0..7
  const int tile_m = blockIdx.y;                             // 0..3124
  const int row = tile_m * 16 + l16;
  const int col = tile_n * 16 + l16;

  v8f acc = {};
  acc = wmma_accum_k128(agg + (size_t)row * D, 1.0f, Wrel, D, col, half, acc);
  acc = wmma_accum_k128(hin + (size_t)row * D, 1.0f, Wroot, D, col, half, acc);

  const float bias = brel[col];
#pragma unroll
  for (int v = 0; v < 8; ++v) {
    const int m = tile_m * 16 + v + half * 8;  // C/D layout: M = v + 8*half
    hout[(size_t)m * D + col] = fmaxf(acc[v] + bias, 0.0f);
  }
}

// One wave per node: accumulate h row into per-graph sums + count.
__global__ void __launch_bounds__(256) pool_nodes(const float* __restrict__ h,
                                                  const int* __restrict__ batch,
                                                  float* __restrict__ sums,
                                                  float* __restrict__ counts) {
  const int gid = blockIdx.x * blockDim.x + threadIdx.x;
  const int node = gid >> 5;
  const int lane = gid & 31;
  if (node >= N_NODES) return;
  const int g = batch[node];
  const float4 v = reinterpret_cast<const float4*>(h + (size_t)node * D)[lane];
  float* o = sums + (size_t)g * D + lane * 4;
  atomicAdd(o + 0, v.x);
  atomicAdd(o + 1, v.y);
  atomicAdd(o + 2, v.z);
  atomicAdd(o + 3, v.w);
  if (lane == 0) atomicAdd(&counts[g], 1.0f);
}

// hidden = (sums / max(counts,1)) @ W1 + b1 ; mean folded into A load scale.
// grid = (2, 8), block = (32, 4).
__global__ void __launch_bounds__(128) head_gemm1(
    const float* __restrict__ sums, const float* __restrict__ counts,
    const float* __restrict__ W1, const float* __restrict__ b1,
    float* __restrict__ hidden) {
  const int lane = threadIdx.x;
  const int half = lane >> 4;
  const int l16 = lane & 15;
  const int tile_n = threadIdx.y + blockDim.y * blockIdx.x;
  const int tile_m = blockIdx.y;
  const int row = tile_m * 16 + l16;
  const int col = tile_n * 16 + l16;

  const float rinv = 1.0f / fmaxf(counts[row], 1.0f);
  v8f acc = {};
  acc = wmma_accum_k128(sums + (size_t)row * D, rinv, W1, D, col, half, acc);

  const float bias = b1[col];
#pragma unroll
  for (int v = 0; v < 8; ++v) {
    const int m = tile_m * 16 + v + half * 8;
    hidden[(size_t)m * D + col] = acc[v] + bias;
  }
}

// out[128,16] = hidden @ W2 + b2 ; one wave per 16-row tile. grid = 8, block = 32.
__global__ void __launch_bounds__(32) head_gemm2(
    const float* __restrict__ hidden, const float* __restrict__ W2,
    const float* __restrict__ b2, float* __restrict__ out) {
  const int lane = threadIdx.x;
  const int half = lane >> 4;
  const int l16 = lane & 15;
  const int tile_m = blockIdx.x;
  const int row = tile_m * 16 + l16;

  v8f acc = {};
  acc = wmma_accum_k128(hidden + (size_t)row * D, 1.0f, W2, OUT_DIM, l16, half,
                        acc);

  const float bias = b2[l16];
#pragma unroll
  for (int v = 0; v < 8; ++v) {
    const int m = tile_m * 16 + v + half * 8;
    out[(size_t)m * OUT_DIM + l16] = acc[v] + bias;
  }
}

extern "C" void kernel_launch(void* const* d_in, const int* in_sizes, int n_in,
                              void* d_out, int out_size, void* d_ws,
                              size_t ws_size, hipStream_t stream) {
  const float* x = (const float*)d_in[0];
  const int* edge = (const int*)d_in[1];
  const int* batch = (const int*)d_in[2];
  const float* Wrel = (const float*)d_in[3];
  const float* brel = (const float*)d_in[4];
  const float* Wroot = (const float*)d_in[5];
  const float* W1 = (const float*)d_in[6];
  const float* b1 = (const float*)d_in[7];
  const float* W2 = (const float*)d_in[8];
  const float* b2 = (const float*)d_in[9];
  float* out = (float*)d_out;

  const int* src = edge;
  const int* dst = edge + N_EDGES;

  float* ws = (float*)d_ws;
  float* hA = ws;
  float* hB = hA + (size_t)N_NODES * D;
  float* agg = hB + (size_t)N_NODES * D;
  float* sums = agg + (size_t)N_NODES * D;
  float* counts = sums + (size_t)N_GRAPHS * D;
  float* hidden = counts + N_GRAPHS;

  const dim3 gemm_block(32, 4);
  const dim3 gemm_grid(D / 64, N_NODES / 16);  // (2, 3125)
  const int scatter_blocks = (N_EDGES * 32 + 255) / 256;

  const float* hin = x;
  float* hout = hA;
  for (int l = 0; l < N_LAYERS; ++l) {
    hipMemsetAsync(agg, 0, (size_t)N_NODES * D * sizeof(float), stream);
    scatter_edges<<<scatter_blocks, 256, 0, stream>>>(hin, src, dst, agg);
    graphconv_gemm<<<gemm_grid, gemm_block, 0, stream>>>(
        agg, hin, Wrel + (size_t)l * D * D, brel + (size_t)l * D,
        Wroot + (size_t)l * D * D, hout);
    hin = hout;
    hout = (hout == hA) ? hB : hA;
  }

  hipMemsetAsync(sums, 0, ((size_t)N_GRAPHS * D + N_GRAPHS) * sizeof(float),
                 stream);
  const int pool_blocks = (N_NODES * 32 + 255) / 256;
  pool_nodes<<<pool_blocks, 256, 0, stream>>>(hin, batch, sums, counts);

  head_gemm1<<<dim3(2, N_GRAPHS / 16), gemm_block, 0, stream>>>(sums, counts,
                                                                W1, b1, hidden);
  head_gemm2<<<N_GRAPHS / 16, 32, 0, stream>>>(hidden, W2, b2, out);
}